// FakeRouter_62878321214299
// MI455X (gfx1250) — compile-verified
//
#include <hip/hip_runtime.h>

typedef __attribute__((ext_vector_type(2))) float v2f;
typedef __attribute__((ext_vector_type(8))) float v8f;

#define HDIM   4096
#define NEXP   64
#define TOPK   4
#define ROWS   128            // tokens per block
#define KC     32             // K chunk staged in LDS
#define AST    36             // padded LDS stride in floats (144B: 16B-aligned, bank-spread)
#define SST    68             // scores LDS stride in floats
#define ABUF   (ROWS * AST)   // 4608 floats
#define WBUF   (NEXP * AST)   // 2304 floats
#define BUFSZ  (ABUF + WBUF)  // 6912 floats per buffer
#define NCHUNK (HDIM / KC)    // 128

#if __has_builtin(__builtin_amdgcn_global_load_async_to_lds_b128)
#define USE_ASYNC 1
#else
#define USE_ASYNC 0
#endif

// 2 buffers (async double-buffer) or scores scratch, whichever is larger
static constexpr int SMEM_FLOATS =
    (2 * BUFSZ) > (ROWS * SST) ? (2 * BUFSZ) : (ROWS * SST); // 13824

__device__ __forceinline__ void wait_async_all() {
#if USE_ASYNC
#if __has_builtin(__builtin_amdgcn_s_wait_asynccnt)
    __builtin_amdgcn_s_wait_asynccnt(0);
#else
    asm volatile("s_wait_asynccnt 0" ::: "memory");
#endif
#endif
}

#if USE_ASYNC
// Builtin signature (from clang diagnostic):
//   (__attribute__((vector_size(16))) int __device__* src,
//    ... int __shared__* ldsdst, imm offset, imm cpol)
typedef int v4i_vs __attribute__((vector_size(16)));
typedef __attribute__((address_space(1))) v4i_vs* as1_v4i_p;
typedef __attribute__((address_space(3))) v4i_vs* as3_v4i_p;

// Issue this thread's share of one K-chunk: 4 x b128 for A, 2 x b128 for W.
__device__ __forceinline__ void stage_chunk_async(
    const float* __restrict__ hidden, const float* __restrict__ weight,
    int tokenBase, int kc0, float* As, float* Ws, int srow, int scol)
{
    #pragma unroll
    for (int p = 0; p < 4; ++p) {
        const int r = srow + p * 32;
        const float* g = hidden + (size_t)(tokenBase + r) * HDIM + kc0 + scol;
        float* l = As + r * AST + scol;
        __builtin_amdgcn_global_load_async_to_lds_b128(
            (as1_v4i_p)g, (as3_v4i_p)l, 0, 0);
    }
    #pragma unroll
    for (int p = 0; p < 2; ++p) {
        const int r = srow + p * 32;
        const float* g = weight + (size_t)r * HDIM + kc0 + scol;
        float* l = Ws + r * AST + scol;
        __builtin_amdgcn_global_load_async_to_lds_b128(
            (as1_v4i_p)g, (as3_v4i_p)l, 0, 0);
    }
}
#endif

// 32 WMMAs on one staged chunk: 8 k-steps of K=4, 4 N-tiles.
__device__ __forceinline__ void compute_chunk(
    const float* As, const float* Ws, int w, int mrow, int hi, v8f c[4])
{
    const float* arow = As + (w * 16 + mrow) * AST + 2 * hi;
    const float* brow = Ws + mrow * AST + 2 * hi; // B[k][n] = W[n][k]
    #pragma unroll
    for (int ks = 0; ks < KC / 4; ++ks) {
        v2f a = *(const v2f*)(arow + ks * 4);
        #pragma unroll
        for (int nt = 0; nt < 4; ++nt) {
            v2f b = *(const v2f*)(brow + nt * 16 * AST + ks * 4);
            c[nt] = __builtin_amdgcn_wmma_f32_16x16x4_f32(
                        false, a, false, b, (short)0, c[nt], false, false);
        }
    }
}

__global__ __launch_bounds__(256)
void router_wmma_kernel(const float* __restrict__ hidden,
                        const float* __restrict__ weight,
                        const float* __restrict__ bias,
                        float* __restrict__ out_scores,
                        int* __restrict__ out_idx)
{
    __shared__ __align__(16) float smem[SMEM_FLOATS];
    __shared__ float sbias[NEXP];

    const int t    = threadIdx.x;
    const int w    = t >> 5;          // wave id 0..7
    const int lane = t & 31;
    const int mrow = lane & 15;       // row within 16x16 fragment
    const int hi   = lane >> 4;       // half-wave select
    const int tokenBase = blockIdx.x * ROWS;

    if (t < NEXP) sbias[t] = bias[t];

    v8f c[4];
    #pragma unroll
    for (int nt = 0; nt < 4; ++nt) c[nt] = (v8f)0.0f;

    const int srow = t >> 3;         // staging row 0..31
    const int scol = (t & 7) * 4;    // staging col (float4 granularity)

#if USE_ASYNC
    // -------- async double-buffered pipeline --------
    stage_chunk_async(hidden, weight, tokenBase, 0,
                      smem, smem + ABUF, srow, scol);
    for (int ch = 0; ch < NCHUNK; ++ch) {
        wait_async_all();          // this wave's pending async loads landed
        __syncthreads();           // everyone's landed; prev chunk consumed
        float* curA = smem + (ch & 1) * BUFSZ;
        float* curW = curA + ABUF;
        if (ch + 1 < NCHUNK) {
            float* nxtA = smem + ((ch + 1) & 1) * BUFSZ;
            stage_chunk_async(hidden, weight, tokenBase, (ch + 1) * KC,
                              nxtA, nxtA + ABUF, srow, scol);
        }
        compute_chunk(curA, curW, w, mrow, hi, c);
    }
#else
    // -------- fallback: synchronous staging --------
    float* As = smem;
    float* Ws = smem + ABUF;
    for (int ch = 0; ch < NCHUNK; ++ch) {
        const int kc0 = ch * KC;
        __syncthreads();
        #pragma unroll
        for (int p = 0; p < 4; ++p) {
            const int r = srow + p * 32;
            const float4 v = *(const float4*)(hidden +
                (size_t)(tokenBase + r) * HDIM + kc0 + scol);
            *(float4*)(As + r * AST + scol) = v;
        }
        #pragma unroll
        for (int p = 0; p < 2; ++p) {
            const int r = srow + p * 32;
            const float4 v = *(const float4*)(weight +
                (size_t)r * HDIM + kc0 + scol);
            *(float4*)(Ws + r * AST + scol) = v;
        }
        __syncthreads();
        compute_chunk(As, Ws, w, mrow, hi, c);
    }
#endif

    // ---- dump logits to LDS (aliases tile buffers; barrier first) ----
    __syncthreads();
    #pragma unroll
    for (int nt = 0; nt < 4; ++nt) {
        #pragma unroll
        for (int vv = 0; vv < 8; ++vv) {
            // C/D layout: VGPR vv holds M = vv + 8*hi, N = lane&15 within tile
            smem[(w * 16 + vv + 8 * hi) * SST + nt * 16 + mrow] = c[nt][vv];
        }
    }
    __syncthreads();

    // ---- per-token softmax + top-4 (E=64; 1 thread per token) ----
    if (t < ROWS) {
        const float* x = smem + t * SST;

        float mx = -3.4e38f;
        #pragma unroll 8
        for (int j = 0; j < NEXP; ++j) mx = fmaxf(mx, x[j] + sbias[j]);

        float s = 0.0f;
        #pragma unroll 8
        for (int j = 0; j < NEXP; ++j) s += __expf(x[j] + sbias[j] - mx);
        const float inv = 1.0f / s;

        unsigned long long used = 0ull;
        int   bidx[TOPK];
        float bval[TOPK];
        for (int kk = 0; kk < TOPK; ++kk) {
            float best = -3.4e38f;
            int   bj   = 0;
            for (int j = 0; j < NEXP; ++j) {
                if ((used >> j) & 1ull) continue;
                const float xv = x[j] + sbias[j];
                if (xv > best) { best = xv; bj = j; }   // strict > : lower index wins ties
            }
            used |= (1ull << bj);
            bidx[kk] = bj;
            bval[kk] = __expf(best - mx) * inv;
        }

        const size_t tok = (size_t)tokenBase + t;
        const float4 z = make_float4(0.f, 0.f, 0.f, 0.f);
        #pragma unroll
        for (int jj = 0; jj < NEXP / 4; ++jj)
            *(float4*)(out_scores + tok * NEXP + jj * 4) = z;
        #pragma unroll
        for (int kk = 0; kk < TOPK; ++kk) {
            out_scores[tok * NEXP + bidx[kk]] = bval[kk];
            out_idx[tok * TOPK + kk] = bidx[kk];
        }
    }
}

extern "C" void kernel_launch(void* const* d_in, const int* in_sizes, int n_in,
                              void* d_out, int out_size, void* d_ws, size_t ws_size,
                              hipStream_t stream) {
    const float* hidden = (const float*)d_in[0];
    const float* weight = (const float*)d_in[1];
    const float* bias   = (const float*)d_in[2];

    const int T = in_sizes[0] / HDIM;          // 16384 tokens
    float* out_scores = (float*)d_out;         // [T,64] fp32
    int*   out_idx    = (int*)(out_scores + (size_t)T * NEXP); // [T,4] int32

    dim3 grid(T / ROWS), block(256);
    router_wmma_kernel<<<grid, block, 0, stream>>>(hidden, weight, bias,
                                                   out_scores, out_idx);
}